// Simple_TensorProduct_11390253269623
// MI455X (gfx1250) — compile-verified
//
#include <hip/hip_runtime.h>
#include <hip/hip_bf16.h>

// e3nn-style uvu tensor product: (256x0e+256x1o) x (1x0e+1x1o) -> (256x0e+256x1o)
// Memory-bound streaming kernel for MI455X (gfx1250):
//   - y tile staged to LDS via CDNA5 async copy (global_load_async_to_lds_b128)
//   - per-channel weights resident in VGPRs across the row loop
//   - coalesced b32/b96 x-loads and out-stores, global_prefetch on x stream

#define TP_MUL  256
#define TP_TILE 64   // rows per block

struct F3 { float a, b, c; };

__global__ __launch_bounds__(256) void tp_uvu_kernel(
    const float* __restrict__ x,     // (nrows, 1024)
    const float* __restrict__ y,     // (nrows, 4)
    const float* __restrict__ w,     // (5, 256)
    float* __restrict__ out,         // (nrows, 1024)
    int nrows)
{
    constexpr float S3   = 0.57735026918962576f;   // 1/sqrt(3)
    constexpr float C0   = 0.70710678118654752f;   // sqrt(1/2)
    constexpr float C0S3 = C0 * S3;
    constexpr float C111 = 0.40824829046386302f;   // 1/sqrt(6)

    __shared__ float ybuf[TP_TILE * 4];

    const int u    = threadIdx.x;          // channel 0..255
    const int row0 = blockIdx.x * TP_TILE;
    const int rem  = (nrows - row0 < TP_TILE) ? (nrows - row0) : TP_TILE;

    // ---- CDNA5 async copy: stage y tile (one b128 == one row of y) into LDS ----
    if (u < rem) {
        unsigned lds_off = (unsigned)(size_t)(&ybuf[u * 4]);
        const float* gsrc = y + (size_t)(row0 + u) * 4;
        asm volatile("global_load_async_to_lds_b128 %0, %1, off"
                     :: "v"(lds_off), "v"(gsrc)
                     : "memory");
    }

    // per-channel path weights: resident for the whole row loop
    const float w0 = w[0 * TP_MUL + u];
    const float w1 = w[1 * TP_MUL + u];
    const float w2 = w[2 * TP_MUL + u];
    const float w3 = w[3 * TP_MUL + u];
    const float w4 = w[4 * TP_MUL + u];

    asm volatile("s_wait_asynccnt 0" ::: "memory");
    __syncthreads();

    const float* xrow = x   + (size_t)row0 * 1024;
    float*       orow = out + (size_t)row0 * 1024;

    for (int r = 0; r < rem; ++r) {
        // prefetch the x row 4 iterations ahead (32 lanes x 128B = full 4KB row)
        if (r + 4 < rem && u < 32)
            __builtin_prefetch(xrow + 4 * 1024 + u * 32, 0, 1);

        const float y0  = ybuf[r * 4 + 0];
        const float y10 = ybuf[r * 4 + 1];
        const float y11 = ybuf[r * 4 + 2];
        const float y12 = ybuf[r * 4 + 3];

        const float x0 = xrow[u];
        const F3 xv = *reinterpret_cast<const F3*>(xrow + TP_MUL + 3 * u);
        const float a0 = xv.a, a1 = xv.b, a2 = xv.c;

        // 0e output: sqrt(1/2)*w0*x0*y0 + sqrt(1/2)/sqrt(3)*w1*(x1 . y1)
        const float dot = fmaf(a0, y10, fmaf(a1, y11, a2 * y12));
        const float o0  = fmaf(C0 * w0 * y0, x0, C0S3 * w1 * dot);

        // 1o output: S3*w2*x0*y1 + S3*w3*y0*x1 + (1/sqrt6)*w4*(x1 x y1)
        const float cx = fmaf(a1, y12, -a2 * y11);
        const float cy = fmaf(a2, y10, -a0 * y12);
        const float cz = fmaf(a0, y11, -a1 * y10);

        const float t2 = S3 * w2 * x0;
        const float t3 = S3 * w3 * y0;
        const float t4 = C111 * w4;

        F3 ov;
        ov.a = fmaf(t2, y10, fmaf(t3, a0, t4 * cx));
        ov.b = fmaf(t2, y11, fmaf(t3, a1, t4 * cy));
        ov.c = fmaf(t2, y12, fmaf(t3, a2, t4 * cz));

        orow[u] = o0;
        *reinterpret_cast<F3*>(orow + TP_MUL + 3 * u) = ov;

        xrow += 1024;
        orow += 1024;
    }
}

extern "C" void kernel_launch(void* const* d_in, const int* in_sizes, int n_in,
                              void* d_out, int out_size, void* d_ws, size_t ws_size,
                              hipStream_t stream) {
    const float* x = (const float*)d_in[0];   // (B, 1024)
    const float* y = (const float*)d_in[1];   // (B, 4)
    const float* w = (const float*)d_in[2];   // (1280,)
    float* out = (float*)d_out;

    const int nrows  = in_sizes[0] / (4 * TP_MUL);
    const int ntiles = (nrows + TP_TILE - 1) / TP_TILE;

    tp_uvu_kernel<<<ntiles, 256, 0, stream>>>(x, y, w, out, nrows);
}